// SAGEConv_91225105367498
// MI455X (gfx1250) — compile-verified
//
#include <hip/hip_runtime.h>

#define N_NODES 100000
#define N_EDGES 1600000
#define D 128

typedef __attribute__((ext_vector_type(2))) float v2f;
typedef __attribute__((ext_vector_type(8))) float v8f;

// ---------------------------------------------------------------------------
// Phase 0: zero the workspace (msg accumulator + degree counts).
// ---------------------------------------------------------------------------
__global__ void sage_zero(float* __restrict__ msg, float* __restrict__ deg) {
    size_t i = (size_t)blockIdx.x * blockDim.x + threadIdx.x;
    const size_t total = (size_t)N_NODES * D;
    if (i < total) msg[i] = 0.0f;
    if (i < N_NODES) deg[i] = 0.0f;
}

// ---------------------------------------------------------------------------
// Phase 1: edge scatter. One wave32 per edge; each lane handles 4 features
// (32 lanes * 4 = 128 = D_IN). feat/msg are both L2-resident (51 MB each),
// so these are L2 hardware FP32 atomics (global_atomic_add_f32, no return).
// ---------------------------------------------------------------------------
__global__ __launch_bounds__(256)
void sage_edge_scatter(const float* __restrict__ feat,
                       const int*   __restrict__ src,
                       const int*   __restrict__ dst,
                       float* __restrict__ msg,
                       float* __restrict__ deg) {
    int wave = (int)((blockIdx.x * blockDim.x + threadIdx.x) >> 5);
    int lane = threadIdx.x & 31;
    if (wave >= N_EDGES) return;

    int s = src[wave];
    int d = dst[wave];

    const float4 v = *(const float4*)(feat + (size_t)s * D + lane * 4);
    float* o = msg + (size_t)d * D + lane * 4;
#if defined(__HIP_DEVICE_COMPILE__)
    unsafeAtomicAdd(o + 0, v.x);
    unsafeAtomicAdd(o + 1, v.y);
    unsafeAtomicAdd(o + 2, v.z);
    unsafeAtomicAdd(o + 3, v.w);
    if (lane == 0) unsafeAtomicAdd(deg + d, 1.0f);
#endif
}

// ---------------------------------------------------------------------------
// Phase 2: out = (msg/deg) @ W^T + b + 0.01 via V_WMMA_F32_16X16X4_F32.
// One wave per 16x16 output tile; block = 8 waves = one 16-row stripe
// across all 8 column tiles. K=128 -> 32 fully-unrolled WMMAs per tile.
// Degree normalization fused into the A-fragment load; bias fused into store.
// ---------------------------------------------------------------------------
__global__ __launch_bounds__(256)
void sage_gemm_wmma(const float* __restrict__ msg,
                    const float* __restrict__ deg,
                    const float* __restrict__ W,     // [D_OUT][D_IN] row-major
                    const float* __restrict__ bias,  // [D_OUT]
                    float* __restrict__ out) {
    const int wave = threadIdx.x >> 5;   // 0..7 -> column tile
    const int lane = threadIdx.x & 31;
    const int row0 = blockIdx.x * 16;
    const int col0 = wave * 16;
    const int lr   = lane & 15;
    const int kg   = lane >> 4;          // 0: K pair {0,1}, 1: K pair {2,3}

    const int r = row0 + lr;
    const float rdeg = 1.0f / fmaxf(deg[r], 1.0f);

    const float* __restrict__ arow = msg + (size_t)r * D;          // A: row M=lr
    const float* __restrict__ brow = W   + (size_t)(col0 + lr) * D; // B[k][n]=W[n][k]

    v8f acc = {};
#pragma unroll
    for (int k0 = 0; k0 < D; k0 += 4) {
        v2f a = *(const v2f*)(arow + k0 + 2 * kg);
        a *= rdeg;                                   // fused mean-normalization
        v2f b = *(const v2f*)(brow + k0 + 2 * kg);
        acc = __builtin_amdgcn_wmma_f32_16x16x4_f32(
            /*neg_a=*/false, a, /*neg_b=*/false, b,
            /*c_mod=*/(short)0, acc, /*reuse_a=*/false, /*reuse_b=*/false);
    }

    // D layout: VGPR j -> rows (j + 8*kg), col = col0 + lr
    const float bval = bias[col0 + lr] + 0.01f;
    float* __restrict__ orow = out + (size_t)row0 * D + col0 + lr;
#pragma unroll
    for (int j = 0; j < 8; ++j) {
        orow[(size_t)(j + 8 * kg) * D] = acc[j] + bval;
    }
}

// ---------------------------------------------------------------------------
// Launch: inputs are (feat, W_neigh, b_neigh, src, dst); output f32 [N,128].
// Workspace: msg = N*D floats (51.2 MB), deg = N floats.
// ---------------------------------------------------------------------------
extern "C" void kernel_launch(void* const* d_in, const int* in_sizes, int n_in,
                              void* d_out, int out_size, void* d_ws, size_t ws_size,
                              hipStream_t stream) {
    const float* feat = (const float*)d_in[0];
    const float* W    = (const float*)d_in[1];
    const float* bias = (const float*)d_in[2];
    const int*   src  = (const int*)d_in[3];
    const int*   dst  = (const int*)d_in[4];
    float* out = (float*)d_out;

    float* msg = (float*)d_ws;                       // N_NODES * D
    float* deg = msg + (size_t)N_NODES * D;          // N_NODES

    const size_t total = (size_t)N_NODES * D;
    const int zb = (int)((total + 255) / 256);
    sage_zero<<<zb, 256, 0, stream>>>(msg, deg);

    const int waves_per_block = 256 / 32;
    const int eb = (N_EDGES + waves_per_block - 1) / waves_per_block;
    sage_edge_scatter<<<eb, 256, 0, stream>>>(feat, src, dst, msg, deg);

    sage_gemm_wmma<<<N_NODES / 16, 256, 0, stream>>>(msg, deg, W, bias, out);
}